// RGCN_42064909697807
// MI455X (gfx1250) — compile-verified
//
#include <hip/hip_runtime.h>
#include <hip/hip_bf16.h>
#include <stdint.h>

// RGCN (N_REL=1, edge_type all-zero => relation mask identically 1).
// Layer: out = X@root + b + scatter_mean(X@W, src->dst); ReLU between layers
// is fused into the layer-2 GEMM's A-fragment reads.

typedef float v2f __attribute__((ext_vector_type(2)));
typedef float v8f __attribute__((ext_vector_type(8)));

#define D 128
#define LDS_STRIDE 132   // 128 + 4 pad: column reads hit distinct banks
#define ROWS 64          // rows of X per block = 4 WMMA row-tiles
#define RT 4             // row tiles per wave

// ---------------------------------------------------------------------------
__global__ void rgcn_zero_f32(float* __restrict__ p, int n) {
    int i = blockIdx.x * blockDim.x + threadIdx.x;
    if (i < n) p[i] = 0.0f;
}

__global__ void rgcn_count_deg(const int* __restrict__ edge_index,
                               float* __restrict__ cnt, int E) {
    int e = blockIdx.x * blockDim.x + threadIdx.x;
    if (e < E) {
        int dst = edge_index[E + e];
        unsafeAtomicAdd(&cnt[dst], 1.0f);
    }
}

// ---------------------------------------------------------------------------
// Fused dual GEMM on the fp32 matrix pipe (V_WMMA_F32_16X16X4_F32).
//   outSelf = act(X) @ Wr + bias ;  outMsg = act(X) @ Wm   (act=ReLU if RELU)
// Block: 256 threads = 8 waves, 64 rows of X staged in LDS once via
// GLOBAL_LOAD_ASYNC_TO_LDS_B128 (ASYNCcnt). Wave w owns output columns
// [16w,16w+16) for all 4 row tiles and BOTH outputs, so each B fragment
// feeds 4 WMMAs and each A fragment 2.
template <int RELU>
__global__ __launch_bounds__(256)
void rgcn_gemm_dual_wmma(const float* __restrict__ X,     // [N,128]
                         const float* __restrict__ Wr,    // root [128,128]
                         const float* __restrict__ Wm,    // W    [128,128]
                         const float* __restrict__ bias,  // [128]
                         float* __restrict__ outSelf,     // [N,128]
                         float* __restrict__ outMsg,      // [N,128]
                         int N) {
    __shared__ float xs[ROWS * LDS_STRIDE];

    const int t    = threadIdx.x;
    const int row0 = blockIdx.x * ROWS;

    // Async-stage the 64x128 A tile: 2048 float4s, 8 per thread, coalesced.
    // Per-lane LDS destination address lets us keep the padded stride.
    #pragma unroll
    for (int i = 0; i < 8; ++i) {
        int f  = t + i * 256;        // float4 slot 0..2047
        int r  = f >> 5;             // row 0..63
        int c4 = (f & 31) * 4;       // col 0..124
        int gr = row0 + r;
        if (gr < N) {
            uint32_t lds = (uint32_t)(uintptr_t)(&xs[r * LDS_STRIDE + c4]);
            const float* g = X + (size_t)gr * D + c4;
            asm volatile("global_load_async_to_lds_b128 %0, %1, off"
                         :: "v"(lds), "v"(g) : "memory");
        }
        // OOB rows: LDS garbage only feeds output rows we never store.
    }
    asm volatile("s_wait_asynccnt 0x0" ::: "memory");
    __syncthreads();

    const int lane = t & 31;
    const int wave = t >> 5;            // 0..7 -> column tile
    const int c0   = wave * 16;
    const int n    = lane & 15;         // col (and A-row-within-tile) index
    const int kh   = (lane >> 4) * 2;   // K-half select: 0 or 2

    const float bv = bias[c0 + n];
    v8f accS[RT], accM[RT];
    #pragma unroll
    for (int rt = 0; rt < RT; ++rt) {
        accS[rt] = (v8f){ bv, bv, bv, bv, bv, bv, bv, bv };
        accM[rt] = (v8f){ 0.f, 0.f, 0.f, 0.f, 0.f, 0.f, 0.f, 0.f };
    }

    // K = 128 in steps of 4. Each step: 4 B dwords -> 8 WMMAs.
    #pragma unroll
    for (int k = 0; k < D; k += 4) {
        v2f br, bm;   // B 4x16 fragment: lane holds B[kh][c0+n], B[kh+1][c0+n]
        br.x = Wr[(size_t)(k + kh) * D + c0 + n];
        br.y = Wr[(size_t)(k + kh + 1) * D + c0 + n];
        bm.x = Wm[(size_t)(k + kh) * D + c0 + n];
        bm.y = Wm[(size_t)(k + kh + 1) * D + c0 + n];
        #pragma unroll
        for (int rt = 0; rt < RT; ++rt) {
            v2f a;    // A 16x4 fragment: lane holds A[m][kh], A[m][kh+1]
            a.x = xs[(rt * 16 + n) * LDS_STRIDE + k + kh];
            a.y = xs[(rt * 16 + n) * LDS_STRIDE + k + kh + 1];
            if (RELU) { a.x = fmaxf(a.x, 0.f); a.y = fmaxf(a.y, 0.f); }
            accS[rt] = __builtin_amdgcn_wmma_f32_16x16x4_f32(
                false, a, false, br, (short)0, accS[rt], false, false);
            accM[rt] = __builtin_amdgcn_wmma_f32_16x16x4_f32(
                false, a, false, bm, (short)0, accM[rt], false, false);
        }
    }

    // D 16x16 layout: VGPR j holds row (j + 8*(lane>>4)), col = lane&15.
    #pragma unroll
    for (int rt = 0; rt < RT; ++rt) {
        const int rb = row0 + rt * 16 + 8 * (lane >> 4);
        #pragma unroll
        for (int j = 0; j < 8; ++j) {
            int r = rb + j;
            if (r < N) {
                outSelf[(size_t)r * D + c0 + n] = accS[rt][j];
                outMsg [(size_t)r * D + c0 + n] = accM[rt][j];
            }
        }
    }
}

// ---------------------------------------------------------------------------
// Edge scatter with fused mean: out[dst] += H[src] / cnt[dst].
// One wave per edge: 32 lanes x float4 = full 128-float row; coalesced gather
// + native fp32 global atomics.
__global__ __launch_bounds__(256)
void rgcn_scatter_mean(const float* __restrict__ H,
                       const int* __restrict__ edge_index,
                       const float* __restrict__ cnt,
                       float* __restrict__ out, int E) {
    int wave = blockIdx.x * (blockDim.x >> 5) + (threadIdx.x >> 5);
    int lane = threadIdx.x & 31;
    if (wave >= E) return;

    int src = edge_index[wave];
    int dst = edge_index[E + wave];
    float rinv = 1.0f / cnt[dst];   // dst has >=1 edge by construction

    const float4 h = *(const float4*)(H + (size_t)src * D + lane * 4);
    float* o = out + (size_t)dst * D + lane * 4;
    unsafeAtomicAdd(o + 0, h.x * rinv);
    unsafeAtomicAdd(o + 1, h.y * rinv);
    unsafeAtomicAdd(o + 2, h.z * rinv);
    unsafeAtomicAdd(o + 3, h.w * rinv);
}

// ---------------------------------------------------------------------------
extern "C" void kernel_launch(void* const* d_in, const int* in_sizes, int n_in,
                              void* d_out, int out_size, void* d_ws, size_t ws_size,
                              hipStream_t stream) {
    const float* x          = (const float*)d_in[0];
    const int*   edge_index = (const int*)d_in[1];
    // d_in[2] edge_type: all zeros with N_REL==1 -> unused
    const float* W1    = (const float*)d_in[3];  // [1,128,128]
    const float* root1 = (const float*)d_in[4];
    const float* b1    = (const float*)d_in[5];
    const float* W2    = (const float*)d_in[6];
    const float* root2 = (const float*)d_in[7];
    const float* b2    = (const float*)d_in[8];
    float* out = (float*)d_out;

    const int N = in_sizes[0] / D;      // 100000
    const int E = in_sizes[1] / 2;      // 800000

    // Workspace layout: cnt[N] | hbuf[N*128] | obuf[N*128]
    char* ws = (char*)d_ws;
    size_t cnt_bytes = (((size_t)N * sizeof(float)) + 255) & ~(size_t)255;
    float* cnt  = (float*)ws;
    float* hbuf = (float*)(ws + cnt_bytes);
    float* obuf = hbuf + (size_t)N * D;

    const int gemm_blocks = (N + ROWS - 1) / ROWS;

    // In-degrees (cnt must be re-zeroed every call).
    rgcn_zero_f32<<<(N + 255) / 256, 256, 0, stream>>>(cnt, N);
    rgcn_count_deg<<<(E + 255) / 256, 256, 0, stream>>>(edge_index, cnt, E);

    // Layer 1: obuf = x@root1 + b1 ; hbuf = x@W1 ; obuf += mean-scatter(hbuf)
    rgcn_gemm_dual_wmma<0><<<gemm_blocks, 256, 0, stream>>>(
        x, root1, W1, b1, obuf, hbuf, N);
    rgcn_scatter_mean<<<(E + 7) / 8, 256, 0, stream>>>(
        hbuf, edge_index, cnt, obuf, E);

    // Layer 2 (ReLU fused into A-fragment reads):
    //   out = relu(obuf)@root2 + b2 ; hbuf = relu(obuf)@W2 ; out += scatter
    rgcn_gemm_dual_wmma<1><<<gemm_blocks, 256, 0, stream>>>(
        obuf, root2, W2, b2, out, hbuf, N);
    rgcn_scatter_mean<<<(E + 7) / 8, 256, 0, stream>>>(
        hbuf, edge_index, cnt, out, E);
}